// GNN_63299228009070
// MI455X (gfx1250) — compile-verified
//
#include <hip/hip_runtime.h>

typedef __attribute__((ext_vector_type(2))) float v2f;
typedef __attribute__((ext_vector_type(8))) float v8f;
typedef __attribute__((ext_vector_type(4))) unsigned int v4u;
typedef __attribute__((ext_vector_type(8))) int v8i;
typedef __attribute__((ext_vector_type(4))) int v4i;

#define N_NODES 122880
#define N_EDGES 983040

// ---------------------------------------------------------------------------
// Generic fp32 WMMA GEMM (direct from global): C = A[M,K] @ B[K,N] (+bias).
// Used for the two small layer GEMMs (K=64 / K=128, B panels tiny & L2-hot).
// ---------------------------------------------------------------------------
template <int NT>
__global__ __launch_bounds__(256) void wmma_gemm_f32(
    const float* __restrict__ A, const float* __restrict__ B,
    const float* __restrict__ bias, float* __restrict__ C,
    int M, int N, int K, int relu)
{
  const int gthread = blockIdx.x * blockDim.x + threadIdx.x;
  const int wave    = gthread >> 5;
  const int lane    = threadIdx.x & 31;

  const int tilesN = N / (16 * NT);
  const int tm     = wave / tilesN;          // wave-uniform
  const int tn     = wave % tilesN;
  if (tm * 16 >= M) return;                  // uniform per wave -> EXEC all-ones

  const int half = lane >> 4;
  const int l15  = lane & 15;
  const int koff = half * 2;

  const float* Arow = A + (size_t)(tm * 16 + l15) * K;

  v8f acc[NT] = {};

  for (int k = 0; k < K; k += 4) {
    v2f a;
    a.x = Arow[k + koff + 0];
    a.y = Arow[k + koff + 1];
#pragma unroll
    for (int t = 0; t < NT; ++t) {
      const int col = (tn * NT + t) * 16 + l15;
      v2f b;
      b.x = B[(size_t)(k + koff + 0) * N + col];
      b.y = B[(size_t)(k + koff + 1) * N + col];
      acc[t] = __builtin_amdgcn_wmma_f32_16x16x4_f32(
          false, a, false, b, (short)0, acc[t], false, false);
    }
  }

  const int rowC = tm * 16 + half * 8;
#pragma unroll
  for (int t = 0; t < NT; ++t) {
    const int col = (tn * NT + t) * 16 + l15;
    const float bv = bias ? bias[col] : 0.0f;
#pragma unroll
    for (int v = 0; v < 8; ++v) {
      float val = acc[t][v] + bv;
      if (relu) val = fmaxf(val, 0.0f);
      C[(size_t)(rowC + v) * N + col] = val;
    }
  }
}

// ---------------------------------------------------------------------------
// FC GEMM with Tensor Data Mover: block = 8 waves = 128x64 output tile.
// B panels (KCx64 fp32) are DMA'd to LDS via tensor_load_to_lds and
// double-buffered (TENSORcnt); WMMAs feed from ds_load.
// ---------------------------------------------------------------------------
#define KC 32

__global__ __launch_bounds__(256) void wmma_gemm_fc_tdm(
    const float* __restrict__ A, const float* __restrict__ B,
    const float* __restrict__ bias, float* __restrict__ C,
    int M, int N, int K)
{
  __shared__ float ldsB[2][KC * 64];

  const int lane  = threadIdx.x & 31;
  const int wave  = threadIdx.x >> 5;             // 0..7 -> M sub-tile
  const int nblk  = N / 64;
  const int bm    = blockIdx.x / nblk;
  const int bn    = blockIdx.x % nblk;
  const int rowB  = (bm * 8 + wave) * 16;
  const int colB  = bn * 64;

  const int half = lane >> 4;
  const int l15  = lane & 15;
  const int koff = half * 2;

  const float* Arow = A + (size_t)(rowB + l15) * K;
  const unsigned lds0 = (unsigned)(uintptr_t)&ldsB[0][0];

  v8f acc[4] = {};
  const int NCHUNK = K / KC;

  // Build + issue one 2-D TDM descriptor (D#) for B chunk c into buffer `buf`.
  auto issue = [&](int c, int buf) {
    const unsigned long long ga =
        (unsigned long long)(uintptr_t)B + ((size_t)c * KC * N + colB) * 4ull;
    v4u g0;
    g0.x = 1u;                                        // count=1, user mode
    g0.y = lds0 + (unsigned)buf * (KC * 64 * 4);      // lds_addr (bytes)
    g0.z = (unsigned)(ga & 0xFFFFFFFFu);              // global_addr[31:0]
    g0.w = (unsigned)((ga >> 32) & 0x1FFFFFFu)        // global_addr[56:32]
         | 0x80000000u;                               // type=2 ("image")
    v8i g1;
    g1[0] = (int)(2u << 16);                          // data_size=4B, no mask
    g1[1] = (int)((unsigned)N << 16);                 // tensor_dim0[15:0]
    g1[2] = (int)((unsigned)N >> 16)                  // tensor_dim0[31:16]
          | (int)(((unsigned)K & 0xFFFFu) << 16);     // tensor_dim1[15:0]
    g1[3] = (int)((unsigned)K >> 16)                  // tensor_dim1[31:16]
          | (int)(64u << 16);                         // tile_dim0 = 64
    g1[4] = KC;                                       // tile_dim1 = KC
    g1[5] = N;                                        // tensor_dim0_stride
    g1[6] = 0;
    g1[7] = 0;
    v4i g2 = {0, 0, 0, 0}, g3 = {0, 0, 0, 0};         // 2-D: groups 2/3 unused
    v8i g4 = {0, 0, 0, 0, 0, 0, 0, 0};                // 6-arg toolchain extra group
    __builtin_amdgcn_tensor_load_to_lds(g0, g1, g2, g3, g4, 0);
  };

  if (wave == 0) issue(0, 0);

  for (int c = 0; c < NCHUNK; ++c) {
    const int cur = c & 1;
    if (wave == 0) __builtin_amdgcn_s_wait_tensorcnt(0);  // chunk c landed
    __syncthreads();
    if (wave == 0 && c + 1 < NCHUNK) issue(c + 1, cur ^ 1);

    const float* Bt = &ldsB[cur][0];
    const int kb = c * KC;
#pragma unroll
    for (int k2 = 0; k2 < KC; k2 += 4) {
      v2f a;
      a.x = Arow[kb + k2 + koff + 0];
      a.y = Arow[kb + k2 + koff + 1];
#pragma unroll
      for (int t = 0; t < 4; ++t) {
        const int col = t * 16 + l15;
        v2f b;
        b.x = Bt[(k2 + koff + 0) * 64 + col];
        b.y = Bt[(k2 + koff + 1) * 64 + col];
        acc[t] = __builtin_amdgcn_wmma_f32_16x16x4_f32(
            false, a, false, b, (short)0, acc[t], false, false);
      }
    }
    __syncthreads();   // done reading buf `cur` before it is re-DMA'd
  }

  const int rowC = rowB + half * 8;
#pragma unroll
  for (int t = 0; t < 4; ++t) {
    const int col = colB + t * 16 + l15;
    const float bv = bias[col];
#pragma unroll
    for (int v = 0; v < 8; ++v)
      C[(size_t)(rowC + v) * N + col] = acc[t][v] + bv;
  }
}

// ---------------------------------------------------------------------------
// Degree / normalization helpers
// ---------------------------------------------------------------------------
__global__ void fill_ones(float* __restrict__ p, int n) {
  int i = blockIdx.x * blockDim.x + threadIdx.x;
  if (i < n) p[i] = 1.0f;
}

__global__ void deg_count(const int* __restrict__ dst, float* __restrict__ deg, int E) {
  int i = blockIdx.x * blockDim.x + threadIdx.x;
  if (i < E) atomicAdd(&deg[dst[i]], 1.0f);
}

__global__ void to_rsqrt(float* __restrict__ p, int n) {
  int i = blockIdx.x * blockDim.x + threadIdx.x;
  if (i < n) p[i] = rsqrtf(p[i]);
}

// ---------------------------------------------------------------------------
// Edge scatter:  out[dst] += h[src] * dinv[src]*dinv[dst]
// ---------------------------------------------------------------------------
__global__ void scatter_agg(const float* __restrict__ h,
                            const int* __restrict__ src,
                            const int* __restrict__ dst,
                            const float* __restrict__ dinv,
                            float* __restrict__ out, int E, int F)
{
  const int quads = F >> 2;
  const long long idx = (long long)blockIdx.x * blockDim.x + threadIdx.x;
  if (idx >= (long long)E * quads) return;
  const int e = (int)(idx / quads);
  const int q = (int)(idx % quads);
  const int s = src[e];
  const int d = dst[e];
  const float norm = dinv[s] * dinv[d];
  const float4 hv = *(const float4*)(h + (size_t)s * F + q * 4);
  float* o = out + (size_t)d * F + q * 4;
  atomicAdd(o + 0, hv.x * norm);
  atomicAdd(o + 1, hv.y * norm);
  atomicAdd(o + 2, hv.z * norm);
  atomicAdd(o + 3, hv.w * norm);
}

// out = relu(out + h * dinv^2 + bias)
__global__ void finalize_relu(const float* __restrict__ h,
                              const float* __restrict__ dinv,
                              const float* __restrict__ bias,
                              float* __restrict__ out, int n, int F)
{
  const long long idx = (long long)blockIdx.x * blockDim.x + threadIdx.x;
  if (idx >= (long long)n * F) return;
  const int i = (int)(idx / F);
  const int f = (int)(idx % F);
  const float di = dinv[i];
  const float v = out[idx] + h[idx] * di * di + bias[f];
  out[idx] = fmaxf(v, 0.0f);
}

// ---------------------------------------------------------------------------
extern "C" void kernel_launch(void* const* d_in, const int* in_sizes, int n_in,
                              void* d_out, int out_size, void* d_ws, size_t ws_size,
                              hipStream_t stream) {
  (void)in_sizes; (void)n_in; (void)out_size; (void)ws_size;

  const float* x   = (const float*)d_in[0];
  const int*   ei  = (const int*)d_in[1];
  const float* W1  = (const float*)d_in[2];
  const float* b1  = (const float*)d_in[3];
  const float* W2  = (const float*)d_in[4];
  const float* b2  = (const float*)d_in[5];
  const float* Wfc = (const float*)d_in[6];
  const float* bfc = (const float*)d_in[7];
  float*       out = (float*)d_out;

  const int n = N_NODES, E = N_EDGES;
  const int* src  = ei;
  const int* dstp = ei + E;

  float* dinv = (float*)d_ws;
  float* bufA = dinv + n;                       // n x 128 (h1, later h2)
  float* bufB = bufA + (size_t)n * 128;         // n x 128 (agg1, later agg2)

  const int BT = 256;

  // --- degree -> dinv ---
  fill_ones<<<(n + BT - 1) / BT, BT, 0, stream>>>(dinv, n);
  deg_count<<<(E + BT - 1) / BT, BT, 0, stream>>>(dstp, dinv, E);
  to_rsqrt<<<(n + BT - 1) / BT, BT, 0, stream>>>(dinv, n);

  // --- layer 1: h1 = x @ W1   [122880x64]@[64x128] ---
  {
    const int M = n, N = 128, K = 64;
    const int waves = (M / 16) * (N / 64);
    wmma_gemm_f32<4><<<waves * 32 / BT, BT, 0, stream>>>(x, W1, nullptr, bufA, M, N, K, 0);
  }
  (void)hipMemsetAsync(bufB, 0, (size_t)n * 128 * sizeof(float), stream);
  {
    const long long tot = (long long)E * (128 / 4);
    scatter_agg<<<(int)((tot + BT - 1) / BT), BT, 0, stream>>>(bufA, src, dstp, dinv, bufB, E, 128);
  }
  finalize_relu<<<(int)(((long long)n * 128 + BT - 1) / BT), BT, 0, stream>>>(bufA, dinv, b1, bufB, n, 128);

  // --- layer 2: h2 = h1r @ W2  [122880x128]@[128x64] ---
  {
    const int M = n, N = 64, K = 128;
    const int waves = (M / 16) * (N / 64);
    wmma_gemm_f32<4><<<waves * 32 / BT, BT, 0, stream>>>(bufB, W2, nullptr, bufA, M, N, K, 0);
  }
  (void)hipMemsetAsync(bufB, 0, (size_t)n * 64 * sizeof(float), stream);
  {
    const long long tot = (long long)E * (64 / 4);
    scatter_agg<<<(int)((tot + BT - 1) / BT), BT, 0, stream>>>(bufA, src, dstp, dinv, bufB, E, 64);
  }
  finalize_relu<<<(int)(((long long)n * 64 + BT - 1) / BT), BT, 0, stream>>>(bufA, dinv, b2, bufB, n, 64);

  // --- FC: [4096x1920] @ [1920x1728] + bfc, TDM-fed LDS double buffering ---
  {
    const int M = 4096, N = 1728, K = 1920;
    const int blocks = (M / 128) * (N / 64);   // 32 * 27 = 864
    wmma_gemm_fc_tdm<<<blocks, BT, 0, stream>>>(bufB, Wfc, bfc, out, M, N, K);
  }
}